// DTNN_4587025072491
// MI455X (gfx1250) — compile-verified
//
#include <hip/hip_runtime.h>
#include <hip/hip_bf16.h>

// CDNA5 / gfx1250, wave32. fp32 WMMA (V_WMMA_F32_16X16X4_F32) for the
// per-edge 64x64 GEMMs. A-operands staged via GLOBAL_LOAD_ASYNC_TO_LDS_B128
// (ASYNCcnt); weights staged in LDS in B-fragment (row-pair) layout so each
// WMMA B operand is one aligned ds_load_b64; f32 atomics for segment sums.

typedef __attribute__((ext_vector_type(2))) float v2f;
typedef __attribute__((ext_vector_type(8))) float v8f;

#define BASIS 64
#define HID   128
#define WSTRIDE 80   // v2f stride per K-row-pair group: 64 cols + 16 pad
                     // (160 floats; 160 mod 64 = 32 -> lane-group halves hit
                     //  complementary banks -> conflict-free b64 reads)
#define SSTRIDE 68   // floats per staged row: 272B (16B-aligned for b128;
                     // 68 mod 64 = 4 -> 16 rows on 16 distinct bank pairs,
                     // +2 for lgrp=1 -> complementary -> conflict-free b64)

__device__ __forceinline__ float fast_tanh(float x) {
  // tanh(x) = 1 - 2/(exp(2x)+1), via hw exp2/rcp. Saturates correctly at +-inf.
  float e = __builtin_amdgcn_exp2f(x * 2.885390081777927f); // 2*log2(e)
  return 1.0f - 2.0f * __builtin_amdgcn_rcpf(e + 1.0f);
}

// Async DMA one 16B chunk: copies 16 bytes from global address gaddr to LDS address lds_addr; tracked by ASYNCcnt.
__device__ __forceinline__ void async_copy_b128(unsigned int lds_addr, const void* gaddr) {
  asm volatile("global_load_async_to_lds_b128 %0, %1, off"
               :: "v"(lds_addr), "v"(gaddr) : "memory");
}
__device__ __forceinline__ void wait_async0() {
  asm volatile("s_wait_asynccnt 0x0" ::: "memory");
}

// ---------------------------------------------------------------------------
// Edge kernel: one wave handles 16 edges; block = 8 waves = 128 edges.
// GEMM1: c_feat = C[src] @ cfW + cfb      (16x64 @ 64x64)
// GEMM2: d_feat = edge_attr @ dfW + dfb   (recomputed each iter; same bytes as caching)
// GEMM3: msg = tanh((c_feat*d_feat) @ fcW)
// scatter: atomicAdd into Cout[dst]  (Cout pre-initialized to Cin => C+agg)
// ---------------------------------------------------------------------------
__global__ __launch_bounds__(256)
void dtnn_edge_wmma(const float* __restrict__ Cin, float* __restrict__ Cout,
                    const int* __restrict__ src, const int* __restrict__ dst,
                    const float* __restrict__ edge_attr,
                    const float* __restrict__ cfW, const float* __restrict__ cfb,
                    const float* __restrict__ dfW, const float* __restrict__ dfb,
                    const float* __restrict__ fcW, int nEdges)
{
  // Weights in B-fragment layout: pair g=kk>>1 at column n holds
  // (W[2g][n], W[2g+1][n]) -> one aligned v2f per WMMA B operand.
  __shared__ v2f s_cfWp[32 * WSTRIDE];
  __shared__ v2f s_dfWp[32 * WSTRIDE];
  __shared__ v2f s_fcWp[32 * WSTRIDE];
  __shared__ float s_cfb[BASIS];
  __shared__ float s_dfb[BASIS];
  // Per-wave staging: async gather target for C[src] rows (later reused for p),
  // and for edge_attr rows. 16 rows x SSTRIDE floats each.
  __shared__ __align__(16) float s_stgC[8][16 * SSTRIDE];
  __shared__ __align__(16) float s_stgE[8][16 * SSTRIDE];

  const int tid = threadIdx.x;
  {
    float* cf = (float*)s_cfWp;
    float* df = (float*)s_dfWp;
    float* ff = (float*)s_fcWp;
    for (int i = tid; i < BASIS * BASIS; i += 256) {
      const int kk = i >> 6, n = i & 63;
      const int idx = ((kk >> 1) * WSTRIDE + n) * 2 + (kk & 1);
      cf[idx] = cfW[i];
      df[idx] = dfW[i];
      ff[idx] = fcW[i];
    }
  }
  if (tid < BASIS) { s_cfb[tid] = cfb[tid]; s_dfb[tid] = dfb[tid]; }
  __syncthreads();

  const int wave = tid >> 5;
  const int lane = tid & 31;
  const int lrow = lane & 15;   // A row / B,C,D column within 16-tile
  const int lgrp = lane >> 4;   // K/M half-select
  const int q    = lane & 15;   // 16B chunk index within a 256B row
  const int h    = lane >> 4;   // row half-select for staging

  const int tile = blockIdx.x * 8 + wave;
  const int e0 = tile * 16;
  if (e0 + 16 > nEdges) return;          // wave-uniform: EXEC stays all-ones

  float* __restrict__ stgC = &s_stgC[wave][0];
  float* __restrict__ stgE = &s_stgE[wave][0];
  const unsigned int ldsC = (unsigned int)(size_t)stgC;
  const unsigned int ldsE = (unsigned int)(size_t)stgE;

  // ---- async-stage the 16 C[src] rows and 16 edge_attr rows into LDS ----
  // Each half-wave copies one full 256B row per step as 16 coalesced 16B chunks.
#pragma unroll
  for (int i = 0; i < 8; ++i) {
    const int row = 2 * i + h;
    const int s   = src[e0 + row];
    const float* gC = Cin + (size_t)s * BASIS + q * 4;
    const float* gE = edge_attr + (size_t)(e0 + row) * BASIS + q * 4;
    const unsigned int loff = (unsigned int)(row * (SSTRIDE * 4) + q * 16);
    async_copy_b128(ldsC + loff, gC);
    async_copy_b128(ldsE + loff, gE);
  }

  int drow[8];
#pragma unroll
  for (int r = 0; r < 8; ++r) drow[r] = dst[e0 + r + 8 * lgrp];

  wait_async0();   // all staged rows visible in LDS

  // ---- A fragments (f32 A 16x4 layout: lane holds row lrow,
  //      cols kbase + 2*lgrp + {0,1}) via conflict-free ds_load_b64 ----
  v2f aC[16], aE[16];
#pragma unroll
  for (int k = 0; k < 16; ++k) {
    const int off = lrow * SSTRIDE + 4 * k + 2 * lgrp;
    aC[k] = *(const v2f*)(stgC + off);
    aE[k] = *(const v2f*)(stgE + off);
  }

  // ---- GEMM1 & GEMM2 interleaved; p = (c_feat)*(d_feat) -> LDS (reuses stgC) ----
#pragma unroll
  for (int nt = 0; nt < 4; ++nt) {
    v8f c = {0.f, 0.f, 0.f, 0.f, 0.f, 0.f, 0.f, 0.f};
    v8f d = {0.f, 0.f, 0.f, 0.f, 0.f, 0.f, 0.f, 0.f};
    const int bn = nt * 16 + lrow;
#pragma unroll
    for (int k = 0; k < 16; ++k) {
      const int gbase = (2 * k + lgrp) * WSTRIDE + bn;
      const v2f bc = s_cfWp[gbase];
      const v2f bd = s_dfWp[gbase];
      c = __builtin_amdgcn_wmma_f32_16x16x4_f32(false, aC[k], false, bc,
                                                (short)0, c, false, false);
      d = __builtin_amdgcn_wmma_f32_16x16x4_f32(false, aE[k], false, bd,
                                                (short)0, d, false, false);
    }
    const float cb = s_cfb[bn];
    const float db = s_dfb[bn];
#pragma unroll
    for (int r = 0; r < 8; ++r) {
      // C/D layout: vgpr r, lane -> element (m = r + 8*lgrp, n = bn).
      // Same-wave LDS ops are in-order: these writes follow the aC reads above.
      const int m = r + 8 * lgrp;
      stgC[m * SSTRIDE + bn] = (c[r] + cb) * (d[r] + db);
    }
  }

  // ---- GEMM3: msg = tanh(p @ fcW); scatter-add to Cout[dst] ----
  v2f aP[16];
#pragma unroll
  for (int k = 0; k < 16; ++k) {
    aP[k] = *(const v2f*)(stgC + lrow * SSTRIDE + 4 * k + 2 * lgrp);
  }
#pragma unroll
  for (int nt = 0; nt < 4; ++nt) {
    v8f m = {0.f, 0.f, 0.f, 0.f, 0.f, 0.f, 0.f, 0.f};
    const int bn = nt * 16 + lrow;
#pragma unroll
    for (int k = 0; k < 16; ++k) {
      const v2f bf = s_fcWp[(2 * k + lgrp) * WSTRIDE + bn];
      m = __builtin_amdgcn_wmma_f32_16x16x4_f32(false, aP[k], false, bf,
                                                (short)0, m, false, false);
    }
#pragma unroll
    for (int r = 0; r < 8; ++r) {
      atomicAdd(&Cout[(size_t)drow[r] * BASIS + bn], fast_tanh(m[r]));
    }
  }
}

// ---------------------------------------------------------------------------
// Scalar tail for edges not covered by full 16-edge tiles (E % 16).
// One block of 64 threads per edge; thread = output feature.
// ---------------------------------------------------------------------------
__global__ void dtnn_edge_tail(const float* __restrict__ Cin, float* __restrict__ Cout,
                               const int* __restrict__ src, const int* __restrict__ dst,
                               const float* __restrict__ edge_attr,
                               const float* __restrict__ cfW, const float* __restrict__ cfb,
                               const float* __restrict__ dfW, const float* __restrict__ dfb,
                               const float* __restrict__ fcW, int eStart, int nEdges)
{
  const int e = eStart + blockIdx.x;
  if (e >= nEdges) return;
  const int n = threadIdx.x;                 // 0..63
  __shared__ float spv[BASIS];
  const float* crow = Cin + (size_t)src[e] * BASIS;
  const float* erow = edge_attr + (size_t)e * BASIS;
  float cf = cfb[n], df = dfb[n];
  for (int k = 0; k < BASIS; ++k) {
    cf += crow[k] * cfW[k * BASIS + n];
    df += erow[k] * dfW[k * BASIS + n];
  }
  spv[n] = cf * df;
  __syncthreads();
  float m = 0.f;
  for (int k = 0; k < BASIS; ++k) m += spv[k] * fcW[k * BASIS + n];
  atomicAdd(&Cout[(size_t)dst[e] * BASIS + n], fast_tanh(m));
}

// ---------------------------------------------------------------------------
// Readout: h = tanh(C@W1+b1)@W2+b2, pooled by batch via atomics. One thread/node.
// ---------------------------------------------------------------------------
__global__ __launch_bounds__(256)
void dtnn_readout(const float* __restrict__ C, const int* __restrict__ batch,
                  const float* __restrict__ W1, const float* __restrict__ b1,
                  const float* __restrict__ W2, const float* __restrict__ b2,
                  float* __restrict__ out, int nNodes)
{
  __shared__ float s_W1[BASIS * HID];
  __shared__ float s_W2[HID * 4];
  __shared__ float s_b1[HID];
  const int tid = threadIdx.x;
  for (int i = tid; i < BASIS * HID; i += 256) s_W1[i] = W1[i];
  for (int i = tid; i < HID * 4; i += 256) s_W2[i] = W2[i];
  if (tid < HID) s_b1[tid] = b1[tid];
  __syncthreads();

  const int node = blockIdx.x * 256 + tid;
  if (node >= nNodes) return;

  float c[BASIS];
#pragma unroll
  for (int k = 0; k < BASIS; ++k) c[k] = C[(size_t)node * BASIS + k];

  float a0 = b2[0], a1 = b2[1], a2 = b2[2], a3 = b2[3];
  for (int j = 0; j < HID; ++j) {
    float h = s_b1[j];
#pragma unroll
    for (int k = 0; k < BASIS; ++k) h += c[k] * s_W1[k * HID + j];
    h = fast_tanh(h);
    a0 += h * s_W2[j * 4 + 0];
    a1 += h * s_W2[j * 4 + 1];
    a2 += h * s_W2[j * 4 + 2];
    a3 += h * s_W2[j * 4 + 3];
  }
  const int g = batch[node];
  atomicAdd(&out[g * 4 + 0], a0);
  atomicAdd(&out[g * 4 + 1], a1);
  atomicAdd(&out[g * 4 + 2], a2);
  atomicAdd(&out[g * 4 + 3], a3);
}

// ---------------------------------------------------------------------------
// Small utility kernels
// ---------------------------------------------------------------------------
__global__ void dtnn_init_C(const int* __restrict__ Z, const float* __restrict__ embed,
                            float* __restrict__ C, int total)
{
  const int i = blockIdx.x * blockDim.x + threadIdx.x;
  if (i < total) C[i] = embed[Z[i >> 6] * BASIS + (i & 63)];
}

__global__ void dtnn_copy4(const float4* __restrict__ in, float4* __restrict__ out, int n4)
{
  const int i = blockIdx.x * blockDim.x + threadIdx.x;
  if (i < n4) out[i] = in[i];
}

__global__ void dtnn_zero(float* __restrict__ p, int n)
{
  const int i = blockIdx.x * blockDim.x + threadIdx.x;
  if (i < n) p[i] = 0.0f;
}

// ---------------------------------------------------------------------------
extern "C" void kernel_launch(void* const* d_in, const int* in_sizes, int n_in,
                              void* d_out, int out_size, void* d_ws, size_t ws_size,
                              hipStream_t stream)
{
  const int*   Z         = (const int*)d_in[0];
  const int*   ei        = (const int*)d_in[1];   // [2, E]: row0=src, row1=dst
  const float* edge_attr = (const float*)d_in[2];
  const int*   batch     = (const int*)d_in[3];
  const float* embed     = (const float*)d_in[4];
  const float* cfW       = (const float*)d_in[5];
  const float* cfb       = (const float*)d_in[6];
  const float* dfW       = (const float*)d_in[7];
  const float* dfb       = (const float*)d_in[8];
  const float* fcW       = (const float*)d_in[9];
  const float* W1        = (const float*)d_in[10];
  const float* b1        = (const float*)d_in[11];
  const float* W2        = (const float*)d_in[12];
  const float* b2        = (const float*)d_in[13];

  const int N = in_sizes[0];          // 50000 nodes
  const int E = in_sizes[1] / 2;      // 800000 edges
  const int* src = ei;
  const int* dst = ei + E;

  float* C0 = (float*)d_ws;                       // [N,64]
  float* C1 = C0 + (size_t)N * BASIS;             // [N,64]  (2*12.8 MB of ws)

  const int totalC = N * BASIS;
  dtnn_init_C<<<(totalC + 255) / 256, 256, 0, stream>>>(Z, embed, C0, totalC);
  dtnn_zero<<<(out_size + 255) / 256, 256, 0, stream>>>((float*)d_out, out_size);

  const int tiles   = E / 16;
  const int wblocks = (tiles + 7) / 8;
  const int rem     = E % 16;        // 0 for E=800000
  const int n4      = totalC / 4;

  for (int t = 0; t < 3; ++t) {
    // Cout starts as a copy of Cin, so scattered atomics produce C + agg.
    dtnn_copy4<<<(n4 + 255) / 256, 256, 0, stream>>>((const float4*)C0, (float4*)C1, n4);
    if (wblocks > 0)
      dtnn_edge_wmma<<<wblocks, 256, 0, stream>>>(C0, C1, src, dst, edge_attr,
                                                  cfW, cfb, dfW, dfb, fcW, E);
    if (rem > 0)
      dtnn_edge_tail<<<rem, 64, 0, stream>>>(C0, C1, src, dst, edge_attr,
                                             cfW, cfb, dfW, dfb, fcW, E - rem, E);
    float* tmp = C0; C0 = C1; C1 = tmp;
  }

  dtnn_readout<<<(N + 255) / 256, 256, 0, stream>>>(C0, batch, W1, b1, W2, b2,
                                                    (float*)d_out, N);
}